// HungarianMatcher_75007308857463
// MI455X (gfx1250) — compile-verified
//
#include <hip/hip_runtime.h>

// HungarianMatcher cost matrix: C[n, j] = L1(pred[n], tgt[j]) - clip(GIoU, -1, 1)
//   pred: [9600, 4] f32 (xyxy), tgt: [1600, 4] f32, out: [9600, 1600] f32
// Output-store bound floor: 61.4 MB @ 23.3 TB/s ~= 2.6 us; VALU ~430M ops is the
// practical limit. Inputs staged via CDNA5 async global->LDS DMA (ASYNCcnt),
// reused via LDS broadcast; stores packed to b128.

constexpr int BSZ     = 32;
constexpr int Q       = 300;
constexpr int T       = 50;
constexpr int NPRED   = BSZ * Q;   // 9600
constexpr int MTGT    = BSZ * T;   // 1600
constexpr int PTILE   = 16;        // pred rows per block
constexpr int THREADS = 256;       // 8 waves (wave32)
constexpr int JV      = 4;         // targets per thread per sweep (b128 stores)

__device__ __forceinline__ float clamp01(float x) {
    return __builtin_fminf(__builtin_fmaxf(x, 0.0f), 1.0f);
}

__global__ __launch_bounds__(THREADS)
void giou_cost_kernel(const float4* __restrict__ pred,
                      const float4* __restrict__ tgt,
                      float* __restrict__ out)
{
    __shared__ float4 sTgt[MTGT];       // 25.6 KB raw target boxes
    __shared__ float4 sPredRaw[PTILE];  // raw pred boxes (L1 term)
    __shared__ float4 sPredC[PTILE];    // clamped pred boxes (GIoU)
    __shared__ float  sArea1[PTILE];    // pred areas

    const int tid   = threadIdx.x;
    const int pbase = blockIdx.x * PTILE;

    // ---- Stage all 1600 target boxes into LDS via CDNA5 async DMA ----
    // 1600 = 6*256 + 64: 6 unconditional issues + 1 predicated tail.
    #pragma unroll
    for (int i = 0; i < MTGT / THREADS; ++i) {
        const int j = i * THREADS + tid;
        unsigned lds = (unsigned)(size_t)(&sTgt[j]);        // low 32b of flat LDS addr == LDS offset
        unsigned go  = (unsigned)(j * (int)sizeof(float4)); // byte offset, GVS mode
        asm volatile("global_load_async_to_lds_b128 %0, %1, %2"
                     :: "v"(lds), "v"(go), "s"(tgt) : "memory");
    }
    if (tid < (MTGT % THREADS)) {
        const int j = (MTGT / THREADS) * THREADS + tid;
        unsigned lds = (unsigned)(size_t)(&sTgt[j]);
        unsigned go  = (unsigned)(j * (int)sizeof(float4));
        asm volatile("global_load_async_to_lds_b128 %0, %1, %2"
                     :: "v"(lds), "v"(go), "s"(tgt) : "memory");
    }
    // ---- Stage this block's 16 pred boxes ----
    if (tid < PTILE) {
        unsigned lds = (unsigned)(size_t)(&sPredRaw[tid]);
        unsigned go  = (unsigned)((pbase + tid) * (int)sizeof(float4));
        asm volatile("global_load_async_to_lds_b128 %0, %1, %2"
                     :: "v"(lds), "v"(go), "s"(pred) : "memory");
    }
    asm volatile("s_wait_asynccnt 0x0" ::: "memory");
    __syncthreads();

    // ---- Precompute clamped preds + areas once per block ----
    if (tid < PTILE) {
        float4 a = sPredRaw[tid];
        float4 c = make_float4(clamp01(a.x), clamp01(a.y), clamp01(a.z), clamp01(a.w));
        sPredC[tid] = c;
        sArea1[tid] = (c.z - c.x) * (c.w - c.y);
    }
    __syncthreads();

    // ---- Main loop: each thread owns 4 consecutive targets per sweep ----
    // Sweep 0: j0 = 4*tid        (covers 0..1023, all 256 threads)
    // Sweep 1: j0 = 1024 + 4*tid (covers 1024..1599, threads 0..143)
    for (int k = 0; k < 2; ++k) {
        const int j0 = k * THREADS * JV + tid * JV;
        if (j0 >= MTGT) break;

        // Load + preprocess 4 targets into registers
        float4 b[JV];
        float  bcx[JV], bcy[JV], bcz[JV], bcw[JV], area2[JV];
        #pragma unroll
        for (int v = 0; v < JV; ++v) {
            b[v]   = sTgt[j0 + v];
            bcx[v] = clamp01(b[v].x);
            bcy[v] = clamp01(b[v].y);
            bcz[v] = clamp01(b[v].z);
            bcw[v] = clamp01(b[v].w);
            area2[v] = (bcz[v] - bcx[v]) * (bcw[v] - bcy[v]);
        }

        #pragma unroll 4
        for (int p = 0; p < PTILE; ++p) {
            const float4 a = sPredRaw[p];   // LDS broadcast reads
            const float4 c = sPredC[p];
            const float area1 = sArea1[p];

            float4 cost4;
            float* cp = &cost4.x;
            #pragma unroll
            for (int v = 0; v < JV; ++v) {
                // L1 cdist on raw coords
                float l1 = __builtin_fabsf(a.x - b[v].x) + __builtin_fabsf(a.y - b[v].y)
                         + __builtin_fabsf(a.z - b[v].z) + __builtin_fabsf(a.w - b[v].w);

                // intersection
                float iw = __builtin_fminf(c.z, bcz[v]) - __builtin_fmaxf(c.x, bcx[v]);
                float ih = __builtin_fminf(c.w, bcw[v]) - __builtin_fmaxf(c.y, bcy[v]);
                iw = __builtin_fmaxf(iw, 0.0f);
                ih = __builtin_fmaxf(ih, 0.0f);
                const float inter = iw * ih;
                const float uni   = area1 + area2[v] - inter;

                // enclosing box
                float ew = __builtin_fmaxf(c.z, bcz[v]) - __builtin_fminf(c.x, bcx[v]);
                float eh = __builtin_fmaxf(c.w, bcw[v]) - __builtin_fminf(c.y, bcy[v]);
                ew = __builtin_fmaxf(ew, 0.0f);
                eh = __builtin_fmaxf(eh, 0.0f);
                const float enc = ew * eh;

                // v_rcp_f32: rcp(0)=+inf, 0*inf=NaN -> matches reference 0/0 NaN
                const float iou = inter * __builtin_amdgcn_rcpf(uni);
                float giou = iou - (enc - uni) * __builtin_amdgcn_rcpf(enc);

                // NaN-propagating clip(giou, -1, 1)
                giou = (giou < -1.0f) ? -1.0f : giou;
                giou = (giou >  1.0f) ?  1.0f : giou;

                float cost = l1 - giou;                 // W_BBOX = W_GIOU = 1
                cp[v] = (cost != cost) ? 1e5f : cost;   // NaN -> 1e5
            }
            // 16B-aligned (MTGT and j0 are multiples of 4): global_store_b128
            *(float4*)(out + (size_t)(pbase + p) * MTGT + j0) = cost4;
        }
    }
}

extern "C" void kernel_launch(void* const* d_in, const int* in_sizes, int n_in,
                              void* d_out, int out_size, void* d_ws, size_t ws_size,
                              hipStream_t stream) {
    const float4* pred = (const float4*)d_in[0];  // [9600] float4
    const float4* tgt  = (const float4*)d_in[1];  // [1600] float4
    float* out = (float*)d_out;                   // [9600 * 1600] f32

    const int nblocks = NPRED / PTILE;            // 600
    giou_cost_kernel<<<dim3(nblocks), dim3(THREADS), 0, stream>>>(pred, tgt, out);
}